// BaseSSM_55344948576350
// MI455X (gfx1250) — compile-verified
//
#include <hip/hip_runtime.h>
#include <cstdint>
#include <cstddef>

// Problem constants (match reference)
#define L_SEQ   8192
#define ED      1024
#define SN      16
#define SN2     (SN / 2)              // state pairs for packed fp32 math

// Tiling
#define NCHUNK  64                    // chunks along sequence
#define TCHUNK  (L_SEQ / NCHUNK)      // 128 timesteps per chunk
#define CB      256                   // channels per block (== blockDim.x)
#define TSUB    16                    // timesteps per LDS subtile
#define NSUB    (TCHUNK / TSUB)       // 8 subtiles per chunk
#define TILE_F32   (TSUB * CB)        // 4096 floats = 16 KB
#define TILE_BYTES (TILE_F32 * 4)

typedef float v2f __attribute__((ext_vector_type(2)));

// ---------------------------------------------------------------------------
// CDNA5 async global->LDS DMA (tracked by ASYNCcnt). GVS mode:
// mem addr = SADDR(SGPR64) + VADDR(i32); VDST = LDS byte address relative to
// the workgroup LDS base (dynamic smem starts at offset 0).
// ---------------------------------------------------------------------------
__device__ __forceinline__ void async_b128_to_lds(uint32_t lds_byte_off,
                                                  const float* sbase,
                                                  uint32_t g_byte_off) {
  asm volatile("global_load_async_to_lds_b128 %0, %1, %2"
               :
               : "v"(lds_byte_off), "v"(g_byte_off), "s"(sbase)
               : "memory");
}

// ---------------------------------------------------------------------------
// Phase 0: discretization. idx = d*16 + n over D*N = 16384 threads.
// ---------------------------------------------------------------------------
__global__ __launch_bounds__(256) void ssm_setup(
    const float* __restrict__ A_log, const float* __restrict__ B,
    const float* __restrict__ dt,
    float* __restrict__ abar, float* __restrict__ bbar,
    float* __restrict__ decay) {
  int idx = blockIdx.x * 256 + threadIdx.x;   // [0, ED*SN)
  int d = idx >> 4;
  float a   = -expf(A_log[idx]);              // strictly negative
  float dte = expf(dt[d]);
  float ab  = expf(a * dte);
  float bb  = (ab - 1.0f) / a * B[idx];
  float dec = ab;                             // ab^(2^7) = ab^TCHUNK
#pragma unroll
  for (int i = 0; i < 7; ++i) dec *= dec;
  abar[idx]  = ab;
  bbar[idx]  = bb;
  decay[idx] = dec;
}

// ---------------------------------------------------------------------------
// Phase 2: inter-chunk prefix scan. One thread per (d,n) state element.
// ---------------------------------------------------------------------------
__global__ __launch_bounds__(256) void ssm_chunkscan(
    const float* __restrict__ decay, const float* __restrict__ hlocal,
    float* __restrict__ hinit) {
  int idx = blockIdx.x * 256 + threadIdx.x;   // [0, ED*SN)
  float dec = decay[idx];
  float h = 0.0f;
  for (int c = 0; c < NCHUNK; ++c) {
    hinit[(size_t)c * (ED * SN) + idx] = h;
    h = __builtin_fmaf(dec, h, hlocal[(size_t)c * (ED * SN) + idx]);
  }
}

// ---------------------------------------------------------------------------
// Phases 1 & 3: intra-chunk scan, double-buffered async LDS streaming,
// packed-fp32 (v_pk_fma_f32 / v_pk_mul_f32) state math.
// Block = 256 threads (8 wave32), one channel per thread.
// ---------------------------------------------------------------------------
template <bool WRITE_Y>
__global__ __launch_bounds__(CB) void ssm_scan(
    const float* __restrict__ x,
    const float* __restrict__ abar_g, const float* __restrict__ bbar_g,
    const float* __restrict__ Cg, const float* __restrict__ Dg,
    const float* __restrict__ hinit,   // read when WRITE_Y
    float* __restrict__ hlocal,        // written when !WRITE_Y
    float* __restrict__ y)             // written when WRITE_Y
{
  extern __shared__ float smem[];      // 2 * TILE_F32 floats (32 KB)
  const int tid   = threadIdx.x;
  const int chunk = blockIdx.x;
  const int d     = blockIdx.y * CB + tid;

  // Per-channel constants as 8 x float2 (packed VOP3P operands).
  v2f ab2[SN2], bb2[SN2], cv2[SN2], h2[SN2];
  {
    const v2f* a2 = reinterpret_cast<const v2f*>(abar_g + (size_t)d * SN);
    const v2f* b2 = reinterpret_cast<const v2f*>(bbar_g + (size_t)d * SN);
#pragma unroll
    for (int q = 0; q < SN2; ++q) { ab2[q] = a2[q]; bb2[q] = b2[q]; }
  }
  float Dd = 0.0f;
  if constexpr (WRITE_Y) {
    const v2f* c2 = reinterpret_cast<const v2f*>(Cg + (size_t)d * SN);
    const v2f* h02 =
        reinterpret_cast<const v2f*>(hinit + (size_t)chunk * ED * SN + (size_t)d * SN);
#pragma unroll
    for (int q = 0; q < SN2; ++q) { cv2[q] = c2[q]; h2[q] = h02[q]; }
    Dd = Dg[d];
  } else {
#pragma unroll
    for (int q = 0; q < SN2; ++q) h2[q] = (v2f)(0.0f);
    (void)cv2;
  }

  // x tile base for this (chunk, channel-block): rows of ED floats.
  const float* xbase = x + (size_t)chunk * TCHUNK * ED + (size_t)blockIdx.y * CB;

  // Issue one 16x256 subtile: 1024 b128 DMAs, 4 per thread (4 per wave -> ASYNCcnt +4).
  auto issue = [&](int s) {
    const float* gb = xbase + (size_t)s * TSUB * ED;
    uint32_t bufoff = (uint32_t)(s & 1) * TILE_BYTES;
#pragma unroll
    for (int k = 0; k < 4; ++k) {
      int i   = tid + k * CB;
      int row = i >> 6;          // 0..15
      int cg  = i & 63;          // 0..63 groups of 4 floats
      uint32_t goff = (uint32_t)(row * ED + cg * 4) * 4u;          // bytes
      uint32_t loff = bufoff + (uint32_t)(row * CB + cg * 4) * 4u; // bytes
      async_b128_to_lds(loff, gb, goff);
    }
  };

  issue(0);
  for (int s = 0; s < NSUB; ++s) {
    if (s + 1 < NSUB) {
      issue(s + 1);                                   // overlap next tile's DMA
      asm volatile("s_wait_asynccnt 4" ::: "memory"); // tile s complete (in-order)
    } else {
      asm volatile("s_wait_asynccnt 0" ::: "memory");
    }
    __syncthreads();                                  // all waves' DMA visible

    const float* tile = smem + (s & 1) * TILE_F32;
#pragma unroll
    for (int tr = 0; tr < TSUB; ++tr) {
      float xv = tile[tr * CB + tid];                 // ds_load_b32, conflict-free
      v2f xv2 = { xv, xv };
      if constexpr (WRITE_Y) {
        v2f acc2 = { __builtin_fmaf(xv, Dd, 0.0f), 0.0f };
#pragma unroll
        for (int q = 0; q < SN2; ++q) {
          h2[q] = __builtin_elementwise_fma(ab2[q], h2[q], bb2[q] * xv2); // pk_mul+pk_fma
          acc2  = __builtin_elementwise_fma(h2[q], cv2[q], acc2);         // pk_fma
        }
        y[(size_t)(chunk * TCHUNK + s * TSUB + tr) * ED + d] = acc2.x + acc2.y;
      } else {
#pragma unroll
        for (int q = 0; q < SN2; ++q)
          h2[q] = __builtin_elementwise_fma(ab2[q], h2[q], bb2[q] * xv2);
      }
    }
    __syncthreads();   // protect buffer before it is re-filled
  }

  if constexpr (!WRITE_Y) {
    v2f* o = reinterpret_cast<v2f*>(hlocal + (size_t)chunk * ED * SN + (size_t)d * SN);
#pragma unroll
    for (int q = 0; q < SN2; ++q) o[q] = h2[q];
  }
}

// ---------------------------------------------------------------------------
// Launch: setup -> local chunk scans -> inter-chunk scan -> final scan + y.
// Workspace: abar/bbar/decay (3*64KB) + hlocal (4MB) + hinit (4MB) ~ 8.2 MB.
// x (32MB) stays L2-resident (192MB) between the two scan passes, so HBM
// traffic ~ 64MB; packed fp32 math addresses the VALU bound.
// ---------------------------------------------------------------------------
extern "C" void kernel_launch(void* const* d_in, const int* in_sizes, int n_in,
                              void* d_out, int out_size, void* d_ws, size_t ws_size,
                              hipStream_t stream) {
  const float* x     = (const float*)d_in[0];
  const float* A_log = (const float*)d_in[1];
  const float* B     = (const float*)d_in[2];
  const float* C     = (const float*)d_in[3];
  const float* D     = (const float*)d_in[4];
  const float* dt    = (const float*)d_in[5];
  float* out = (float*)d_out;

  float* abar   = (float*)d_ws;
  float* bbar   = abar + ED * SN;
  float* decay  = bbar + ED * SN;
  float* hlocal = decay + ED * SN;
  float* hinit  = hlocal + (size_t)NCHUNK * ED * SN;

  ssm_setup<<<(ED * SN) / 256, 256, 0, stream>>>(A_log, B, dt, abar, bbar, decay);

  dim3 grid(NCHUNK, ED / CB);
  ssm_scan<false><<<grid, CB, 2 * TILE_BYTES, stream>>>(
      x, abar, bbar, C, D, nullptr, hlocal, nullptr);

  ssm_chunkscan<<<(ED * SN) / 256, 256, 0, stream>>>(decay, hlocal, hinit);

  ssm_scan<true><<<grid, CB, 2 * TILE_BYTES, stream>>>(
      x, abar, bbar, C, D, hinit, nullptr, out);
}